// RVQQuantizer_71150428226058
// MI455X (gfx1250) — compile-verified
//
#include <hip/hip_runtime.h>
#include <stdint.h>

// ---------------- problem constants ----------------
#define G       2
#define KCODES  1024
#define EDIM    256
#define BATCH   8
#define TLEN    4096
#define CDIM    512
#define NT      (BATCH*TLEN)        // 32768 rows per group
#define ROWS    64                  // rows per workgroup tile
#define CHUNK   64                  // codes per LDS chunk
#define NCHUNK  (KCODES/CHUNK)      // 16
#define PITCH   260                 // LDS row pitch (floats): conflict-free b64 frags, 1040B = 65*16
#define A_OFF   0
#define CB0_OFF (ROWS*PITCH)            // 16640
#define CB1_OFF (2*ROWS*PITCH)          // 33280
#define KEY_OFFF (3*ROWS*PITCH)         // 49920 (u64 keys: 64 rows -> 128 floats, 8B aligned)
#define RED_OFFF (KEY_OFFF + 2*ROWS)    // 50048
#define LDS_FLOATS (RED_OFFF + 16)      // 50064 floats = 200,256 bytes (1 WG / WGP)

typedef __attribute__((ext_vector_type(2))) float v2f;
typedef __attribute__((ext_vector_type(8))) float v8f;

// ---- order-preserving (dist, idx) packing: u64 min == (min dist, then min idx) ----
__device__ __forceinline__ unsigned long long pack_key(float d, unsigned idx) {
    unsigned u = __float_as_uint(d);
    u = (u & 0x80000000u) ? ~u : (u | 0x80000000u);
    return ((unsigned long long)u << 32) | (unsigned long long)idx;
}
__device__ __forceinline__ float unpack_dist(unsigned long long k) {
    unsigned u = (unsigned)(k >> 32);
    u = (u & 0x80000000u) ? (u & 0x7FFFFFFFu) : ~u;
    return __uint_as_float(u);
}

// ---- CDNA5 async global->LDS staging (ASYNCcnt-tracked, bypasses VGPRs) ----
// Copies 64 codes x 256 floats (64 KiB) into LDS at buf_off, 16B per lane per issue.
__device__ __forceinline__ void stage_chunk_async(float* lds, int buf_off,
                                                  const float* __restrict__ src, int tid) {
    #pragma unroll
    for (int it = 0; it < 16; ++it) {
        int flat = (it * 256 + tid) * 4;          // float index within chunk
        int code = flat >> 8;
        int d    = flat & (EDIM - 1);
        uint32_t lofs = (uint32_t)(uintptr_t)&lds[buf_off + code * PITCH + d];
        const float* g = src + flat;              // 16B aligned (flat % 4 == 0)
        asm volatile("global_load_async_to_lds_b128 %0, %1, off"
                     :: "v"(lofs), "v"(g) : "memory");
    }
}
__device__ __forceinline__ void wait_async0() {
    asm volatile("s_wait_asynccnt 0" ::: "memory");
}

// ---------------- codebook norms + loss-accumulator init ----------------
// norms layout: [step(2)][g(2)][k(1024)]
__global__ void __launch_bounds__(256) rvq_norms(const float* __restrict__ cb1,
                                                 const float* __restrict__ cb2,
                                                 float* __restrict__ norms,
                                                 float* __restrict__ lossacc) {
    if (blockIdx.x == 0 && threadIdx.x < 2) lossacc[threadIdx.x] = 0.0f;
    const int wave = threadIdx.x >> 5;
    const int lane = threadIdx.x & 31;
    const int code = blockIdx.x * 8 + wave;            // 0..4095
    const float* src = (code < G*KCODES) ? cb1 : cb2;
    const int local = code & (G*KCODES - 1);           // g*K + k
    const float* row = src + (size_t)local * EDIM;
    float s = 0.0f;
    for (int i = lane; i < EDIM; i += 32) { float v = row[i]; s += v * v; }
    #pragma unroll
    for (int m = 16; m; m >>= 1) s += __shfl_xor(s, m, 32);
    if (lane == 0) norms[code] = s;
}

// ---------------- one RVQ step (WMMA distance GEMM + argmin) ----------------
__global__ void __launch_bounds__(256) rvq_step(
        const float* __restrict__ x,      // (B,C,T)
        const float* __restrict__ cb,     // (G,K,E) codebook for this step
        const float* __restrict__ norms,  // (G*K) ||cb||^2 for this step
        float* __restrict__ resid,        // ws (G,NT,E): step0 writes, step1 reads
        float* __restrict__ qout,         // (B,C,T): step1 writes q1+q2
        float* __restrict__ idx_out,      // 2*NT floats for this step
        float* __restrict__ lossacc,      // 1 float for this step
        int step)
{
    extern __shared__ float lds[];
    unsigned long long* keys = (unsigned long long*)&lds[KEY_OFFF];

    const int tid = threadIdx.x;
    const int g  = blockIdx.y;
    const int n0 = blockIdx.x * ROWS;
    const int b  = n0 >> 12;            // / TLEN
    const int t0 = n0 & (TLEN - 1);

    if (tid < ROWS) keys[tid] = ~0ull;

    const float* cbg = cb + (size_t)g * KCODES * EDIM;
    const float* nrm = norms + g * KCODES;

    // kick off async staging of code chunk 0 into buffer 0 immediately
    stage_chunk_async(lds, CB0_OFF, cbg, tid);

    // ---- stage A tile (64 rows x 256 dims) scaled by -2 (exact); accumulate ||row||^2 ----
    float sq = 0.0f;
    if (step == 0) {
        // x[b, g*E + d, t0 + row]: consecutive tid -> consecutive t (coalesced)
        const float* xb = x + ((size_t)b * CDIM + (size_t)g * EDIM) * TLEN + t0;
        for (int i = tid; i < ROWS * EDIM; i += 256) {
            int row = i & (ROWS - 1);
            int d   = i >> 6;
            float v = xb[(size_t)d * TLEN + row];
            lds[A_OFF + row * PITCH + d] = -2.0f * v;
            sq += v * v;
        }
    } else {
        // residual[g, n0+row, d]: consecutive tid -> consecutive d (coalesced)
        const float* rp = resid + ((size_t)g * NT + n0) * EDIM;
        for (int i = tid; i < ROWS * EDIM; i += 256) {
            int d   = i & (EDIM - 1);
            int row = i >> 8;
            float v = rp[(size_t)row * EDIM + d];
            lds[A_OFF + row * PITCH + d] = -2.0f * v;
            sq += v * v;
        }
    }

    const int lane = tid & 31;
    const int w    = tid >> 5;
    const int rb_  = w & 3;     // row block (16 rows)
    const int h    = w >> 2;    // code half (32 codes)
    const int lo   = lane & 15;
    const int hi   = lane >> 4;

    unsigned long long runkey[8];
    #pragma unroll
    for (int v = 0; v < 8; ++v) runkey[v] = ~0ull;

    const float* ap = &lds[A_OFF + (rb_ * 16 + lo) * PITCH + 2 * hi];
    const int bfrag = (h * 32 + lo) * PITCH + 2 * hi;

    wait_async0();        // chunk 0 resident
    __syncthreads();      // + A tile visible to all waves

    for (int c = 0; c < NCHUNK; ++c) {
        const int k0 = c * CHUNK;
        const int cur_off = (c & 1) ? CB1_OFF : CB0_OFF;
        const int nxt_off = (c & 1) ? CB0_OFF : CB1_OFF;

        // prefetch next chunk into the other buffer (async, overlaps the GEMM)
        if (c + 1 < NCHUNK)
            stage_chunk_async(lds, nxt_off, cbg + (size_t)(k0 + CHUNK) * EDIM, tid);

        const float* bp0 = &lds[cur_off + bfrag];
        const float* bp1 = bp0 + 16 * PITCH;

        v8f acc0 = {};
        v8f acc1 = {};
        #pragma unroll 8
        for (int d = 0; d < EDIM; d += 4) {
            v2f a  = *(const v2f*)(ap  + d);
            v2f b0 = *(const v2f*)(bp0 + d);
            v2f b1 = *(const v2f*)(bp1 + d);
            acc0 = __builtin_amdgcn_wmma_f32_16x16x4_f32(false, a, false, b0,
                                                         (short)0, acc0, false, false);
            acc1 = __builtin_amdgcn_wmma_f32_16x16x4_f32(false, a, false, b1,
                                                         (short)0, acc1, false, false);
        }

        // partial dist = ||cb||^2 - 2 x.cb ; reduce over 16 columns per row
        const float n0v = nrm[k0 + h * 32 + lo];
        const float n1v = nrm[k0 + h * 32 + 16 + lo];
        #pragma unroll
        for (int v = 0; v < 8; ++v) {
            unsigned long long key0 = pack_key(acc0[v] + n0v, (unsigned)(k0 + h * 32 + lo));
            unsigned long long key1 = pack_key(acc1[v] + n1v, (unsigned)(k0 + h * 32 + 16 + lo));
            unsigned long long kk = key0 < key1 ? key0 : key1;
            #pragma unroll
            for (int m = 1; m <= 8; m <<= 1) {
                unsigned long long o = __shfl_xor(kk, m, 16);
                if (o < kk) kk = o;
            }
            if (kk < runkey[v]) runkey[v] = kk;
        }

        wait_async0();    // next buffer fully in LDS (and this wave's issues retired)
        __syncthreads();  // all waves done reading cur before it is restaged
    }

    // merge per-wave running minima into per-row LDS keys (ds_min_u64)
    if (lo == 0) {
        #pragma unroll
        for (int v = 0; v < 8; ++v)
            atomicMin(&keys[rb_ * 16 + hi * 8 + v], runkey[v]);
    }
    __syncthreads();

    // ---- loss: sum(min full dist) = sum(min partial) + sum(||row||^2) ----
    float dsum = (tid < ROWS) ? unpack_dist(keys[tid]) : 0.0f;
    float tot = sq + dsum;
    #pragma unroll
    for (int m = 16; m; m >>= 1) tot += __shfl_xor(tot, m, 32);
    if (lane == 0) lds[RED_OFFF + w] = tot;
    __syncthreads();
    if (tid == 0) {
        float s = 0.0f;
        for (int i = 0; i < 8; ++i) s += lds[RED_OFFF + i];
        atomicAdd(lossacc, s);
    }

    // ---- index output (as float; values < 1024 are exact) ----
    if (tid < ROWS)
        idx_out[(size_t)g * NT + n0 + tid] =
            (float)(unsigned)(keys[tid] & 0xFFFFFFFFull);

    if (step == 0) {
        // residual = x - cb[g, idx] ; A_lds holds -2x -> x = -0.5*A (exact)
        float* rw = resid + ((size_t)g * NT + n0) * EDIM;
        for (int i = tid; i < ROWS * EDIM; i += 256) {
            int d   = i & (EDIM - 1);
            int row = i >> 8;
            unsigned idx = (unsigned)(keys[row] & 0xFFFFFFFFull);
            float q = cbg[(size_t)idx * EDIM + d];
            rw[(size_t)row * EDIM + d] = -0.5f * lds[A_OFF + row * PITCH + d] - q;
        }
    } else {
        // stage zq2 tile (coalesced gather: one code row per iteration)
        for (int i = tid; i < ROWS * EDIM; i += 256) {
            int d   = i & (EDIM - 1);
            int row = i >> 8;
            unsigned idx = (unsigned)(keys[row] & 0xFFFFFFFFull);
            lds[CB0_OFF + row * PITCH + d] = cbg[(size_t)idx * EDIM + d];
        }
        __syncthreads();
        // out[b, g*E+d, t0+row] = x - r2 + zq2 (coalesced along t); r2 = -0.5*A_lds
        const float* xb = x    + ((size_t)b * CDIM + (size_t)g * EDIM) * TLEN + t0;
        float*       ob = qout + ((size_t)b * CDIM + (size_t)g * EDIM) * TLEN + t0;
        for (int i = tid; i < ROWS * EDIM; i += 256) {
            int row = i & (ROWS - 1);
            int d   = i >> 6;
            float xv = xb[(size_t)d * TLEN + row];
            float r  = -0.5f * lds[A_OFF + row * PITCH + d];
            float q  = lds[CB0_OFF + row * PITCH + d];
            ob[(size_t)d * TLEN + row] = xv - r + q;
        }
    }
}

// ---------------- final scalar loss ----------------
__global__ void rvq_finalize(const float* __restrict__ lossacc, float* __restrict__ loss_out) {
    if (threadIdx.x == 0) {
        // loss = (1.25*S1/BTC + 1.25*S2/BTC) / 2
        *loss_out = 1.25f * (lossacc[0] + lossacc[1]) * (1.0f / (2.0f * 16777216.0f));
    }
}

extern "C" void kernel_launch(void* const* d_in, const int* in_sizes, int n_in,
                              void* d_out, int out_size, void* d_ws, size_t ws_size,
                              hipStream_t stream) {
    const float* x   = (const float*)d_in[0];
    const float* cb1 = (const float*)d_in[1];
    const float* cb2 = (const float*)d_in[2];

    float* out      = (float*)d_out;
    float* loss_out = out + (size_t)BATCH * CDIM * TLEN;   // after 16,777,216 floats
    float* idx_out  = loss_out + 1;                        // 4*NT floats

    char*  ws      = (char*)d_ws;
    float* resid   = (float*)ws;                                        // G*NT*E floats (64 MiB)
    float* norms   = (float*)(ws + (size_t)G * NT * EDIM * 4);          // 2*G*K floats
    float* lossacc = (float*)(ws + (size_t)G * NT * EDIM * 4
                                 + (size_t)2 * G * KCODES * 4);         // 2 floats

    // codebook norms (4096 codes, 8 per WG) + loss accumulator init
    rvq_norms<<<(2 * G * KCODES) / 8, 256, 0, stream>>>(cb1, cb2, norms, lossacc);

    dim3 grid(NT / ROWS, G);
    size_t shmem = (size_t)LDS_FLOATS * 4;

    rvq_step<<<grid, 256, shmem, stream>>>(x, cb1, norms,             resid, out,
                                           idx_out,            lossacc + 0, 0);
    rvq_step<<<grid, 256, shmem, stream>>>(x, cb2, norms + G * KCODES, resid, out,
                                           idx_out + 2 * NT,   lossacc + 1, 1);

    rvq_finalize<<<1, 32, 0, stream>>>(lossacc, loss_out);
}